// GINConv_2997887172726
// MI455X (gfx1250) — compile-verified
//
#include <hip/hip_runtime.h>

// GINConv: agg = x + scatter_add(x[row] -> col); h = relu(agg@W1+b1); out = h@W2+b2
// N=100000 (multiple of 16), E=640000, D=128, fp32 throughout (reference dtype).

#define D 128

typedef __attribute__((ext_vector_type(2))) float v2f;
typedef __attribute__((ext_vector_type(8))) float v8f;

// ---------------------------------------------------------------------------
// Kernel 1: agg[i] = (1+EPS) * x[i]   (EPS == 0 -> plain copy), float4 vectorized
// ---------------------------------------------------------------------------
__global__ __launch_bounds__(256) void gin_init_kernel(const float4* __restrict__ x,
                                                       float4* __restrict__ agg,
                                                       int n4) {
    int i = blockIdx.x * blockDim.x + threadIdx.x;
    if (i < n4) agg[i] = x[i];
}

// ---------------------------------------------------------------------------
// Kernel 2: scatter-add. One wave32 per edge; lane l owns floats [4l, 4l+4).
// row/col are int64 (jnp.int64). 4x global_atomic_add_f32 per lane.
// ---------------------------------------------------------------------------
__global__ __launch_bounds__(256) void gin_scatter_kernel(const float* __restrict__ x,
                                                          const long long* __restrict__ row,
                                                          const long long* __restrict__ col,
                                                          float* agg, int E) {
    int wave = (int)((blockIdx.x * blockDim.x + threadIdx.x) >> 5);
    int lane = threadIdx.x & 31;
    if (wave >= E) return;
    long long r = row[wave];
    long long c = col[wave];
    float4 v = reinterpret_cast<const float4*>(x + r * D)[lane];
    float* dst = agg + c * D + lane * 4;
    __hip_atomic_fetch_add(dst + 0, v.x, __ATOMIC_RELAXED, __HIP_MEMORY_SCOPE_AGENT);
    __hip_atomic_fetch_add(dst + 1, v.y, __ATOMIC_RELAXED, __HIP_MEMORY_SCOPE_AGENT);
    __hip_atomic_fetch_add(dst + 2, v.z, __ATOMIC_RELAXED, __HIP_MEMORY_SCOPE_AGENT);
    __hip_atomic_fetch_add(dst + 3, v.w, __ATOMIC_RELAXED, __HIP_MEMORY_SCOPE_AGENT);
}

// ---------------------------------------------------------------------------
// Kernel 3/4: C[N,128] = act(A[N,128] @ W[128,128] + bias)
// Block = 256 threads = 8 waves. Block covers 16 rows x 128 cols; wave w owns
// the 16x16 tile at columns [16w, 16w+16). K-loop (128, step 4) fully unrolled
// -> 32 x v_wmma_f32_16x16x4_f32 per wave.
//
// Fragment layouts (CDNA5 ISA):
//   A 16x4 : lane M = lane&15; VGPR0/1 hold K = 2*(lane>>4) + {0,1}  -> float2 load
//   B 4x16 : lane N = lane&15; VGPR0/1 hold K = 2*(lane>>4) + {0,1}  -> two b32 loads
//   C/D    : 8 VGPRs; VGPR r -> row (r + 8*(lane>>4)), col = lane&15
//
// BARRIER=true makes in-place C==A safe: every load is consumed by WMMA before
// __syncthreads(), and blocks own disjoint 16-row stripes.
// ---------------------------------------------------------------------------
template <bool RELU, bool BARRIER>
__global__ __launch_bounds__(256) void gin_gemm_kernel(const float* A,
                                                       const float* __restrict__ W,
                                                       const float* __restrict__ bias,
                                                       float* C) {
    const int lane = threadIdx.x & 31;
    const int wave = threadIdx.x >> 5;
    const int l16  = lane & 15;
    const int hi   = lane >> 4;          // 0 or 1: selects K pair / row half
    const int m0   = blockIdx.x * 16;
    const int n0   = wave * 16;

    const float* arow = A + (size_t)(m0 + l16) * D + 2 * hi;   // 8B aligned
    const float* wcol = W + (size_t)(2 * hi) * D + n0 + l16;

    v8f acc = {};
#pragma unroll
    for (int k0 = 0; k0 < D; k0 += 4) {
        v2f a = *reinterpret_cast<const v2f*>(arow + k0);      // global_load_b64
        v2f b;
        b.x = wcol[(size_t)k0 * D];
        b.y = wcol[(size_t)k0 * D + D];
        acc = __builtin_amdgcn_wmma_f32_16x16x4_f32(
            /*neg_a=*/false, a, /*neg_b=*/false, b,
            /*c_mod=*/(short)0, acc, /*reuse_a=*/false, /*reuse_b=*/false);
    }

    const float bv = bias[n0 + l16];
#pragma unroll
    for (int r = 0; r < 8; ++r) {
        float v = acc[r] + bv;
        if (RELU) v = fmaxf(v, 0.0f);
        acc[r] = v;
    }

    if (BARRIER) __syncthreads();        // all block reads of A done -> safe in-place

    const int rb = m0 + hi * 8;
#pragma unroll
    for (int r = 0; r < 8; ++r)
        C[(size_t)(rb + r) * D + n0 + l16] = acc[r];
}

// ---------------------------------------------------------------------------
extern "C" void kernel_launch(void* const* d_in, const int* in_sizes, int n_in,
                              void* d_out, int out_size, void* d_ws, size_t ws_size,
                              hipStream_t stream) {
    const float*     x  = (const float*)d_in[0];
    const long long* ei = (const long long*)d_in[1];  // [2, E] int64
    const float*     W1 = (const float*)d_in[2];
    const float*     b1 = (const float*)d_in[3];
    const float*     W2 = (const float*)d_in[4];
    const float*     b2 = (const float*)d_in[5];
    float*           out = (float*)d_out;

    const int N = in_sizes[0] / D;       // 100000 (multiple of 16)
    const int E = in_sizes[1] / 2;       // 640000

    float* agg = (float*)d_ws;           // N*D*4 = 51.2 MB scratch

    // 1) agg = x
    const int n4 = N * D / 4;
    gin_init_kernel<<<(n4 + 255) / 256, 256, 0, stream>>>(
        (const float4*)x, (float4*)agg, n4);

    // 2) agg[col] += x[row]  (one wave per edge, 8 edges per block)
    gin_scatter_kernel<<<(E + 7) / 8, 256, 0, stream>>>(
        x, ei, ei + E, agg, E);

    // 3) h = relu(agg @ W1 + b1) -> d_out
    gin_gemm_kernel<true, false><<<N / 16, 256, 0, stream>>>(agg, W1, b1, out);

    // 4) out = h @ W2 + b2, in place on d_out (barrier-protected)
    gin_gemm_kernel<false, true><<<N / 16, 256, 0, stream>>>(out, W2, b2, out);
}